// WaveletVolume_61993557950731
// MI455X (gfx1250) — compile-verified
//
#include <hip/hip_runtime.h>
#include <hip/hip_bf16.h>

typedef __attribute__((ext_vector_type(2))) float v2f;
typedef __attribute__((ext_vector_type(8))) float v8f;

#define INV_SCALE 0.35355339059327373f   // 1/(2*sqrt(2))
#define NCH 28

// ---------------------------------------------------------------------------
// One level of 3-D inverse Haar DWT as a WMMA GEMM:
//   out(ncells x 8) = coeff(ncells x 8) * H8 * INV_SCALE
// H8[s][o] = (-1)^popc(s&o).  One wave = 16 cells (M), K=8 via two K=4 WMMAs,
// N columns 0..7 = output corner offsets (oz,oy,ox) = (o>>2, (o>>1)&1, o&1).
//
// approx  : addr = c*acs + sp*ass            (sp = (z*D+y)*D+x)
// details : (7,C,D,D,D) -> addr = ((s-1)*C + c)*D^3 + sp
// out     : channels-last (z2,y2,x2,C)
// ---------------------------------------------------------------------------
__global__ __launch_bounds__(128) void idwt_haar_wmma_kernel(
    const float* __restrict__ apx, const float* __restrict__ det,
    float* __restrict__ out, int D, int acs, int ass)
{
    const int wave = (blockIdx.x * blockDim.x + threadIdx.x) >> 5;
    const int lane = threadIdx.x & 31;

    const int D3 = D * D * D;
    const int cellBase = wave * 16;               // x-fastest enumeration
    const int xb = cellBase % D;                  // multiple of 16, D >= 16
    const int y  = (cellBase / D) % D;
    const int z  = (cellBase / (D * D)) % D;
    const int c  = cellBase / D3;

    // ---- A fragments: this lane's cell is (lane & 15); K pair per half-wave
    const int mloc  = lane & 15;
    const int kbase = (lane < 16) ? 0 : 2;        // lanes 0-15: K0,K1 ; 16-31: K2,K3
    const int sp    = (z * D + y) * D + (xb + mloc);

    // coefficient fetch: subband 0 = approx, 1..7 = details
    #define COEFF(S) ((S) == 0 ? apx[c * acs + sp * ass] \
                               : det[(((S) - 1) * NCH + c) * D3 + sp])
    v2f a1, a2;
    a1[0] = COEFF(kbase + 0);
    a1[1] = COEFF(kbase + 1);
    a2[0] = COEFF(kbase + 4);
    a2[1] = COEFF(kbase + 5);
    #undef COEFF

    // ---- B fragments: column n = lane&15 (cols 8..15 zero), same K mapping.
    // Sign = (-1)^popc(s & n), scaled by INV_SCALE.
    const int n = lane & 15;
    v2f b1, b2;
    #pragma unroll
    for (int j = 0; j < 2; ++j) {
        const int s1 = kbase + j;                 // first  WMMA: subbands 0..3
        const int s2 = kbase + j + 4;             // second WMMA: subbands 4..7
        float h1 = (n < 8) ? ((__popc(s1 & n) & 1) ? -INV_SCALE : INV_SCALE) : 0.0f;
        float h2 = (n < 8) ? ((__popc(s2 & n) & 1) ? -INV_SCALE : INV_SCALE) : 0.0f;
        if (j == 0) { b1[0] = h1; b2[0] = h2; } else { b1[1] = h1; b2[1] = h2; }
    }

    // ---- chained WMMA: acc = A1*B1 + A2*B2  (EXEC is all-ones here)
    v8f acc = {};
    acc = __builtin_amdgcn_wmma_f32_16x16x4_f32(false, a1, false, b1,
                                                (short)0, acc, false, false);
    acc = __builtin_amdgcn_wmma_f32_16x16x4_f32(false, a2, false, b2,
                                                (short)0, acc, false, false);

    // ---- scatter: lane with o = lane&15 < 8 holds corner o of 8 cells.
    // D VGPR r = (M = r (+8 for upper half), N = lane&15).
    const int o = lane & 15;
    if (o < 8) {
        const int mb = (lane < 16) ? 0 : 8;
        const int ox = o & 1, oy = (o >> 1) & 1, oz = o >> 2;
        const int R  = 2 * D;
        const int x2 = 2 * (xb + mb) + ox;
        const int y2 = 2 * y + oy;
        const int z2 = 2 * z + oz;
        float* dst = out + ((z2 * R + y2) * R + x2) * NCH + c;
        #pragma unroll
        for (int r = 0; r < 8; ++r)
            dst[r * 2 * NCH] = acc[r];            // x2 advances by 2 per cell
    }
}

// ---------------------------------------------------------------------------
// Trilinear query with zero padding, align_corners=True, bound=1.5.
// One wave per point; lane = channel (lanes 28..31 idle on gather/store).
// vol is channels-last (128,128,128,28).
// ---------------------------------------------------------------------------
__global__ __launch_bounds__(256) void trilinear_query_kernel(
    const float* __restrict__ vol, const float* __restrict__ xyz,
    float* __restrict__ out, int N)
{
    const int wave = (blockIdx.x * blockDim.x + threadIdx.x) >> 5;
    const int lane = threadIdx.x & 31;
    if (wave >= N) return;

    const int R = 128;
    const float px = (xyz[wave * 3 + 0] * (1.0f / 1.5f) + 1.0f) * 0.5f * 127.0f;
    const float py = (xyz[wave * 3 + 1] * (1.0f / 1.5f) + 1.0f) * 0.5f * 127.0f;
    const float pz = (xyz[wave * 3 + 2] * (1.0f / 1.5f) + 1.0f) * 0.5f * 127.0f;

    const int x0 = (int)floorf(px);
    const int y0 = (int)floorf(py);
    const int z0 = (int)floorf(pz);
    const float fx = px - (float)x0;
    const float fy = py - (float)y0;
    const float fz = pz - (float)z0;

    const int c = lane;
    float acc = 0.0f;
    #pragma unroll
    for (int corner = 0; corner < 8; ++corner) {   // dz outer .. dx inner order
        const int dx = corner & 1, dy = (corner >> 1) & 1, dz = corner >> 2;
        const int ix = x0 + dx, iy = y0 + dy, iz = z0 + dz;
        const bool valid = ((unsigned)ix < (unsigned)R) &
                           ((unsigned)iy < (unsigned)R) &
                           ((unsigned)iz < (unsigned)R);
        const float w = (dx ? fx : 1.0f - fx) *
                        (dy ? fy : 1.0f - fy) *
                        (dz ? fz : 1.0f - fz);
        float v = 0.0f;
        if (valid & (c < NCH))
            v = vol[((iz * R + iy) * R + ix) * NCH + c];
        acc += v * w;
    }
    if (c < NCH)
        out[wave * NCH + c] = acc;
}

// ---------------------------------------------------------------------------
extern "C" void kernel_launch(void* const* d_in, const int* in_sizes, int n_in,
                              void* d_out, int out_size, void* d_ws, size_t ws_size,
                              hipStream_t stream)
{
    const float* approx    = (const float*)d_in[0];   // (28,16,16,16)
    const float* details_0 = (const float*)d_in[1];   // (7,28,16,16,16)
    const float* details_1 = (const float*)d_in[2];   // (7,28,32,32,32)
    const float* details_2 = (const float*)d_in[3];   // (7,28,64,64,64)
    const float* xyz       = (const float*)d_in[4];   // (N,3)
    float* out = (float*)d_out;
    const int N = in_sizes[4] / 3;

    // workspace layout (channels-last volumes)
    char* ws = (char*)d_ws;
    float* vol32  = (float*)(ws);                         //  32^3*28*4 = 3.5 MiB
    float* vol64  = (float*)(ws + (size_t)4  * 1024 * 1024);  // 28 MiB
    float* vol128 = (float*)(ws + (size_t)36 * 1024 * 1024);  // 224 MiB

    // level 0: 16 -> 32   (approx is C-major: acs=D^3, ass=1)
    {
        const int D = 16, waves = NCH * D * D * D / 16;
        idwt_haar_wmma_kernel<<<waves / 4, 128, 0, stream>>>(
            approx, details_0, vol32, D, D * D * D, 1);
    }
    // level 1: 32 -> 64   (approx is channels-last: acs=1, ass=C)
    {
        const int D = 32, waves = NCH * D * D * D / 16;
        idwt_haar_wmma_kernel<<<waves / 4, 128, 0, stream>>>(
            vol32, details_1, vol64, D, 1, NCH);
    }
    // level 2: 64 -> 128
    {
        const int D = 64, waves = NCH * D * D * D / 16;
        idwt_haar_wmma_kernel<<<waves / 4, 128, 0, stream>>>(
            vol64, details_2, vol128, D, 1, NCH);
    }
    // trilinear sampling: one wave per point, 8 waves per block
    {
        const int blocks = (N + 7) / 8;
        trilinear_query_kernel<<<blocks, 256, 0, stream>>>(vol128, xyz, out, N);
    }
}